// Attention_38740605010449
// MI455X (gfx1250) — compile-verified
//
#include <hip/hip_runtime.h>

typedef float v2f __attribute__((ext_vector_type(2)));
typedef float v4f __attribute__((ext_vector_type(4)));
typedef float v8f __attribute__((ext_vector_type(8)));

#define Bsz 2
#define Hn  8
#define Sn  2048
#define Dn  64
#define QT  16          // query rows per workgroup
#define NW  8           // waves per workgroup (256 threads)
#define KT  (Sn / 16)   // 128 key tiles of 16
#define SP  2052        // padded LDS row stride (floats): 2052%64=4 -> conflict-free strided reads

// LDS layout (floats):
//   E[16][SP]      : unnormalized exp(scores) tile
//   rowsum[16]     : row sums -> reciprocals
//   obuf[16][64]   : O accumulation
#define E_OFF   0
#define RS_OFF  (QT * SP)
#define OB_OFF  (RS_OFF + 16)
#define SMEM_FLOATS (OB_OFF + QT * Dn)

__global__ __launch_bounds__(256) void
attn_wmma_f32_kernel(const float* __restrict__ q,
                     const float* __restrict__ k,
                     const float* __restrict__ v,
                     const unsigned char* __restrict__ mask,
                     float* __restrict__ attn,
                     float* __restrict__ outv)
{
    extern __shared__ float smem[];
    float* E      = smem + E_OFF;
    float* rowsum = smem + RS_OFF;
    float* obuf   = smem + OB_OFF;

    const int tid  = threadIdx.x;
    const int lane = tid & 31;
    const int wave = tid >> 5;
    const int lo   = lane & 15;   // N (col) index within 16-wide frag
    const int hi   = lane >> 4;   // half-wave select

    const int bh    = blockIdx.x / (Sn / QT);   // consecutive blocks share a head -> K/V L2-resident
    const int qt    = blockIdx.x % (Sn / QT);
    const int b     = bh / Hn;
    const int qbase = qt * QT;

    const float* qp = q + (size_t)bh * Sn * Dn;
    const float* kp = k + (size_t)bh * Sn * Dn;
    const float* vp = v + (size_t)bh * Sn * Dn;
    const unsigned char* mp = mask + (size_t)b * Sn;

    // zero rowsum + obuf (contiguous)
    for (int i = tid; i < 16 + QT * Dn; i += 256) rowsum[i] = 0.0f;
    __syncthreads();

    // ---- load Q fragments, pre-scaled by 1/sqrt(D) (each wave holds a full copy) ----
    // A-frag layout (16x4 f32): lane L -> M = L%16; low half K={0,1}, high half K={2,3}
    const float* qrow = qp + (size_t)(qbase + lo) * Dn;
    v2f qf[16];
#pragma unroll
    for (int c = 0; c < 16; ++c) {
        const float2 t = *(const float2*)(qrow + 4 * c + 2 * hi);
        qf[c].x = t.x * 0.125f;
        qf[c].y = t.y * 0.125f;
    }

    // ---- Phase 1: scores = QK^T, exp, stash to LDS, row sums ----
    float rs[8] = {0.f, 0.f, 0.f, 0.f, 0.f, 0.f, 0.f, 0.f};

    for (int kt = wave; kt < KT; kt += NW) {
        const int   kcol = kt * 16 + lo;
        const float* krow = kp + (size_t)kcol * Dn;
        const bool  mk   = (mp[kcol] != 0);

        v8f acc = {};
#pragma unroll
        for (int c = 0; c < 16; ++c) {
            // B-frag (4x16): vgpr v -> K = v + 2*hi, N = lane%16 ; B[K=d][N] = Key[kcol][d]
            const float2 t = *(const float2*)(krow + 4 * c + 2 * hi);
            v2f bf; bf.x = t.x; bf.y = t.y;
            acc = __builtin_amdgcn_wmma_f32_16x16x4_f32(
                false, qf[c], false, bf, (short)0, acc, false, false);
        }

        // C layout: vgpr r -> M = r + 8*hi, N = lane%16
#pragma unroll
        for (int r = 0; r < 8; ++r) {
            const int m  = r + 8 * hi;
            const int qg = qbase + m;
            float e = (mk || (qg == kcol)) ? 0.0f : __expf(acc[r]);
            E[m * SP + kcol] = e;
            rs[r] += e;
        }
    }

    // reduce row partial sums across the 16 lanes of each half-wave
#pragma unroll
    for (int r = 0; r < 8; ++r) {
        float t = rs[r];
        t += __shfl_xor(t, 1);
        t += __shfl_xor(t, 2);
        t += __shfl_xor(t, 4);
        t += __shfl_xor(t, 8);
        rs[r] = t;
    }
    if (lo == 0) {
#pragma unroll
        for (int r = 0; r < 8; ++r) atomicAdd(&rowsum[8 * hi + r], rs[r]);
    }
    __syncthreads();

    if (tid < 16) rowsum[tid] = 1.0f / rowsum[tid];   // rowsum now holds reciprocals
    __syncthreads();

    // ---- Phase 2a: write normalized attention (coalesced NT 128-bit stores) ----
    float* ao = attn + ((size_t)bh * Sn + qbase) * Sn;
    for (int i = tid; i < QT * (Sn / 4); i += 256) {
        const int row = i >> 9;        // /512 float4-chunks per row
        const int ch  = i & 511;
        v4f val = *(const v4f*)(E + row * SP + ch * 4);
        const float rinv = rowsum[row];
        val.x *= rinv; val.y *= rinv; val.z *= rinv; val.w *= rinv;
        __builtin_nontemporal_store(val, (v4f*)(ao + (size_t)row * Sn + ch * 4));
    }

    // ---- Phase 2b: O = E @ V (unnormalized; scale rows at the end) ----
    v8f accO[4] = {{}, {}, {}, {}};
    const int kstart = wave * (Sn / NW);   // 256-wide K slice per wave
    for (int c = 0; c < (Sn / NW) / 4; ++c) {
        const int kk = kstart + 4 * c;
        // A-frag from LDS E: lane -> M = lo, K = 2*hi + {0,1}
        const float2 ta = *(const float2*)(E + lo * SP + kk + 2 * hi);
        v2f af; af.x = ta.x; af.y = ta.y;
#pragma unroll
        for (int n = 0; n < 4; ++n) {
            v2f bf;
            bf.x = vp[(size_t)(kk + 2 * hi)     * Dn + 16 * n + lo];
            bf.y = vp[(size_t)(kk + 2 * hi + 1) * Dn + 16 * n + lo];
            accO[n] = __builtin_amdgcn_wmma_f32_16x16x4_f32(
                false, af, false, bf, (short)0, accO[n], false, false);
        }
    }
    // cross-wave reduction into LDS
#pragma unroll
    for (int n = 0; n < 4; ++n)
#pragma unroll
        for (int r = 0; r < 8; ++r)
            atomicAdd(&obuf[(r + 8 * hi) * Dn + 16 * n + lo], accO[n][r]);
    __syncthreads();

    // ---- write sum_value: 16x64 floats, one float4 per thread, row-scaled ----
    {
        const int row = tid >> 4;
        const int ch  = tid & 15;
        v4f val = *(const v4f*)(obuf + row * Dn + ch * 4);
        const float rinv = rowsum[row];
        val.x *= rinv; val.y *= rinv; val.z *= rinv; val.w *= rinv;
        *(v4f*)(outv + ((size_t)(bh * Sn) + qbase + row) * Dn + ch * 4) = val;
    }
}

extern "C" void kernel_launch(void* const* d_in, const int* in_sizes, int n_in,
                              void* d_out, int out_size, void* d_ws, size_t ws_size,
                              hipStream_t stream) {
    (void)in_sizes; (void)n_in; (void)d_ws; (void)ws_size; (void)out_size;
    const float*         q  = (const float*)d_in[0];
    const float*         k  = (const float*)d_in[1];
    const float*         v  = (const float*)d_in[2];
    const unsigned char* m  = (const unsigned char*)d_in[3];

    float* attn = (float*)d_out;                               // [B,H,S,S]
    float* outv = (float*)d_out + (size_t)Bsz * Hn * Sn * Sn;  // [B,H,S,D]

    const dim3 grid(Bsz * Hn * (Sn / QT));   // 2048 workgroups
    const dim3 block(256);
    const size_t smem = (size_t)SMEM_FLOATS * sizeof(float);   // ~135 KB dynamic LDS

    attn_wmma_f32_kernel<<<grid, block, smem, stream>>>(q, k, v, m, attn, outv);
}